// ConvolutionRefinement_48206712930878
// MI455X (gfx1250) — compile-verified
//
#include <hip/hip_runtime.h>
#include <math.h>

// ConvolutionRefinement for MI455X (gfx1250, wave32).
// v_gated [B=4,H=16,L=4096,D=128] fp32 -> RMSNorm(D) -> depthwise causal conv
// (K=4 along L) -> SiLU -> +residual.
// Bandwidth-bound: 268MB traffic, ~11.5us floor at 23.3 TB/s. One wave32 per
// 128-row chunk; lane = 4 channels (32 lanes x float4 = D=128) so every row is
// one coalesced global_load_b128 / global_store_b128 per wave.

typedef float v4f __attribute__((ext_vector_type(4)));

#define ROWS    64      // B*H
#define LEN     4096    // L
#define DIM     128     // D
#define CHUNK   128     // L rows per wave  (LEN/CHUNK == 32 == power of two)
#define CPR     32      // chunks per row
#define CPR_SH  5
#define WPB     8       // waves per block (256 threads)
#define PFD     16      // prefetch distance in rows

__device__ __forceinline__ float wave_sum32(float v) {
    v += __shfl_xor(v, 1, 32);
    v += __shfl_xor(v, 2, 32);
    v += __shfl_xor(v, 4, 32);
    v += __shfl_xor(v, 8, 32);
    v += __shfl_xor(v, 16, 32);
    return v;
}

__device__ __forceinline__ float dot4(v4f v) {
    return fmaf(v.w, v.w, fmaf(v.z, v.z, fmaf(v.y, v.y, v.x * v.x)));
}

// 1/sqrt: single v_rsq_f32 (trans unit), no OCML range-fixup expansion.
__device__ __forceinline__ float fast_rsqrt(float a) {
#if __has_builtin(__builtin_amdgcn_rsqf)
    return __builtin_amdgcn_rsqf(a);
#else
    return rsqrtf(a);
#endif
}

// sigmoid(y): prefer gfx1250 V_TANH_F32 (1 trans + 1 mul + 1 fma), else
// explicit v_exp_f32/v_rcp_f32 chain (2 trans + 3 valu).
__device__ __forceinline__ float fast_sigmoid(float y) {
#if __has_builtin(__builtin_amdgcn_tanhf)
    return fmaf(0.5f, __builtin_amdgcn_tanhf(0.5f * y), 0.5f);
#elif __has_builtin(__builtin_amdgcn_exp2f) && __has_builtin(__builtin_amdgcn_rcpf)
    return __builtin_amdgcn_rcpf(1.0f + __builtin_amdgcn_exp2f(y * -1.442695040888963f));
#else
    return 1.0f / (1.0f + __expf(-y));
#endif
}

__global__ __launch_bounds__(256)
void ConvolutionRefinement_48206712930878_kernel(const float* __restrict__ x,
                                                 const float* __restrict__ nw,
                                                 const float* __restrict__ cw,
                                                 const float* __restrict__ cb,
                                                 float* __restrict__ out) {
    const int lane   = threadIdx.x & 31;
    const int waveId = blockIdx.x * WPB + (threadIdx.x >> 5);
    const int row    = waveId >> CPR_SH;     // 0..63 == b*H+h   (grid is exact)
    const int chunk  = waveId & (CPR - 1);

    const int d0 = lane * 4;                 // this lane's 4 channels

    // Per-lane constants: bias and conv taps with norm_weight folded in.
    const v4f nw4 = *(const v4f*)(nw + d0);
    const v4f cb4 = *(const v4f*)(cb + d0);
    const v4f wr0 = *(const v4f*)(cw + (d0 + 0) * 4);   // cw is [D][K=4] row-major
    const v4f wr1 = *(const v4f*)(cw + (d0 + 1) * 4);
    const v4f wr2 = *(const v4f*)(cw + (d0 + 2) * 4);
    const v4f wr3 = *(const v4f*)(cw + (d0 + 3) * 4);
    const v4f w0n = (v4f){wr0.x, wr1.x, wr2.x, wr3.x} * nw4;
    const v4f w1n = (v4f){wr0.y, wr1.y, wr2.y, wr3.y} * nw4;
    const v4f w2n = (v4f){wr0.z, wr1.z, wr2.z, wr3.z} * nw4;
    const v4f w3n = (v4f){wr0.w, wr1.w, wr2.w, wr3.w} * nw4;

    const int l0 = chunk * CHUNK;
    const float* __restrict__ px = x   + (size_t)row * LEN * DIM + (size_t)l0 * DIM + d0;
    float*       __restrict__ po = out + (size_t)row * LEN * DIM + (size_t)l0 * DIM + d0;

    // Causal left context: ring of previous normalized rows (v*r; nw is in taps).
    v4f p1 = (v4f){0.f, 0.f, 0.f, 0.f};
    v4f p2 = p1, p3 = p1;
    if (l0 > 0) {                            // uniform branch; chunk==0 keeps zeros
        v4f a = *(const v4f*)(px - 3 * DIM);
        v4f b = *(const v4f*)(px - 2 * DIM);
        v4f c = *(const v4f*)(px - 1 * DIM);
        float sa = wave_sum32(dot4(a));      // three independent butterfly chains,
        float sb = wave_sum32(dot4(b));      // compiler interleaves them
        float sc = wave_sum32(dot4(c));
        p3 = a * fast_rsqrt(fmaf(sa, 1.0f / (float)DIM, 1e-6f));
        p2 = b * fast_rsqrt(fmaf(sb, 1.0f / (float)DIM, 1e-6f));
        p1 = c * fast_rsqrt(fmaf(sc, 1.0f / (float)DIM, 1e-6f));
    }

    // Main streaming loop: constant-offset loads (i*512B fits 24-bit IOFFSET),
    // 4x unroll -> compiler clauses 4 loads up front (implicit SW pipeline).
#pragma unroll 4
    for (int i = 0; i < CHUNK; ++i) {
        v4f v = *(const v4f*)(px + i * DIM);
        // gfx1250 global_prefetch_b8, unconditional: OOB prefetch is speculative
        // and silently dropped, so no clamp math needed.
        __builtin_prefetch(px + (i + PFD) * DIM, 0, 0);

        // RMSNorm over D=128: per-lane dot4 + wave32 xor-butterfly (ds_swizzle).
        float ss = wave_sum32(dot4(v));
        float r  = fast_rsqrt(fmaf(ss, 1.0f / (float)DIM, 1e-6f));
        v4f  q   = v * r;

        // Depthwise causal conv (taps carry norm_weight):
        // y = bias + w0*xn[l-3] + w1*xn[l-2] + w2*xn[l-1] + w3*xn[l]
        v4f y = cb4 + w0n * p3 + w1n * p2 + w2n * p1 + w3n * q;

        // SiLU + residual (tanh-based sigmoid: 1 trans op per component).
        v4f o;
        o.x = fmaf(y.x, fast_sigmoid(y.x), v.x);
        o.y = fmaf(y.y, fast_sigmoid(y.y), v.y);
        o.z = fmaf(y.z, fast_sigmoid(y.z), v.z);
        o.w = fmaf(y.w, fast_sigmoid(y.w), v.w);

        // Write-once output: non-temporal b128 store keeps input L2-resident
        // (192MB L2 can hold the whole 134MB input).
        __builtin_nontemporal_store(o, (v4f*)(po + i * DIM));

        p3 = p2; p2 = p1; p1 = q;
    }
}

extern "C" void kernel_launch(void* const* d_in, const int* in_sizes, int n_in,
                              void* d_out, int out_size, void* d_ws, size_t ws_size,
                              hipStream_t stream) {
    (void)in_sizes; (void)n_in; (void)out_size; (void)d_ws; (void)ws_size;
    const float* x  = (const float*)d_in[0];   // v_gated [4,16,4096,128]
    const float* nw = (const float*)d_in[1];   // norm_weight [128]
    const float* cw = (const float*)d_in[2];   // conv_weight [128,4]
    const float* cb = (const float*)d_in[3];   // conv_bias [128]
    float* outp = (float*)d_out;

    const int totalWaves = ROWS * CPR;             // 2048 waves
    dim3 block(WPB * 32);                          // 256 threads = 8 waves
    dim3 grid(totalWaves / WPB);                   // 256 blocks, exact
    ConvolutionRefinement_48206712930878_kernel<<<grid, block, 0, stream>>>(x, nw, cw, cb, outp);
}